// LagrangeInterp_25314537243158
// MI455X (gfx1250) — compile-verified
//
#include <hip/hip_runtime.h>
#include <hip/hip_bf16.h>
#include <stdint.h>

// 2D quartic (D=4) Lagrange interpolation on a 64x64 mesh.
// Table: 257x257 f32 = 258 KB -> lives entirely in one WGP's 320 KB LDS,
// staged with CDNA5 GLOBAL_LOAD_ASYNC_TO_LDS_B128 (ASYNCcnt engine).
// Gathers exploit the guaranteed alignment of the 5x5 patch (start index
// == 0 mod 4, row i starts at i mod 4) to use b128/b64/b32 LDS loads:
// 13 ds ops per point instead of 25. Two points per iteration for ILP.

#define NGRID 257
#define NTAB (NGRID * NGRID)             // 66049 floats
#define TAB_BYTES (NTAB * sizeof(float)) // 264196 bytes of LDS
#define NCHUNK (NTAB / 4)                // 16512 full 16-byte chunks + 1 tail float

#if defined(__HIP_DEVICE_COMPILE__) && defined(__gfx1250__)
  #if __has_builtin(__builtin_amdgcn_global_load_async_to_lds_b128)
    #define HAVE_ASYNC_B128 1
  #else
    #define HAVE_ASYNC_B128 0
  #endif
  #if __has_builtin(__builtin_amdgcn_s_wait_asynccnt)
    #define HAVE_WAIT_ASYNC 1
  #else
    #define HAVE_WAIT_ASYNC 0
  #endif
#else
  #define HAVE_ASYNC_B128 0
  #define HAVE_WAIT_ASYNC 0
#endif

#if HAVE_ASYNC_B128
typedef int v4i __attribute__((vector_size(16)));
typedef __attribute__((address_space(1))) v4i* gptr_v4i;  // global (AS1)
typedef __attribute__((address_space(3))) v4i* lptr_v4i;  // LDS (AS3)
#endif

typedef float v4f __attribute__((ext_vector_type(4)));
typedef float v2f __attribute__((ext_vector_type(2)));

// Quartic Lagrange basis on nodes {0,1,2,3,4}; denominators 24,-6,4,-6,24.
__device__ __forceinline__ void basis5(float t, float b[5]) {
  const float d0 = t;
  const float d1 = t - 1.0f;
  const float d2 = t - 2.0f;
  const float d3 = t - 3.0f;
  const float d4 = t - 4.0f;
  const float d01 = d0 * d1;
  const float d12 = d1 * d2;
  const float d23 = d2 * d3;
  const float d34 = d3 * d4;
  b[0] = d12 * d34 * (1.0f / 24.0f);
  b[1] = d0 * d2 * d34 * (-1.0f / 6.0f);
  b[2] = d01 * d34 * 0.25f;
  b[3] = d01 * d2 * d4 * (-1.0f / 6.0f);
  b[4] = d01 * d23 * (1.0f / 24.0f);
}

// Map one (x,y) to its patch base index (== 0 mod 4) and the two bases.
__device__ __forceinline__ int prep_point(float x, float y,
                                          float b0[5], float b1[5]) {
  const float u0 = x * 64.0f;
  const float u1 = y * 64.0f;
  const float c0 = fminf(fmaxf(floorf(u0), 0.0f), 63.0f);
  const float c1 = fminf(fmaxf(floorf(u1), 0.0f), 63.0f);
  basis5((u0 - c0) * 4.0f, b0);
  basis5((u1 - c1) * 4.0f, b1);
  return (((int)c0) * NGRID + (int)c1) << 2;  // 4*(257*c0 + c1)
}

// 5x5 patch gather + bilinear form, using alignment-aware LDS loads.
// base == 0 (mod 4 floats); row i starts at base + i*257 == i (mod 4).
__device__ __forceinline__ float gather_dot(const float* __restrict__ tab,
                                            int base,
                                            const float b0[5],
                                            const float b1[5]) {
  float acc0, acc1, acc2, acc3, acc4;
  {  // row 0: 16B-aligned -> b128 + b32
    const v4f q = *(const v4f*)(tab + base);
    float s = q.x * b1[0];
    s = fmaf(q.y, b1[1], s);
    s = fmaf(q.z, b1[2], s);
    s = fmaf(q.w, b1[3], s);
    s = fmaf(tab[base + 4], b1[4], s);
    acc0 = b0[0] * s;
  }
  {  // row 1 (==1 mod 4): b32 + b64 + b64
    const int r = base + NGRID;
    const float f0 = tab[r];
    const v2f qa = *(const v2f*)(tab + r + 1);
    const v2f qb = *(const v2f*)(tab + r + 3);
    float s = f0 * b1[0];
    s = fmaf(qa.x, b1[1], s);
    s = fmaf(qa.y, b1[2], s);
    s = fmaf(qb.x, b1[3], s);
    s = fmaf(qb.y, b1[4], s);
    acc1 = b0[1] * s;
  }
  {  // row 2 (==2 mod 4): b64 + b64 + b32
    const int r = base + 2 * NGRID;
    const v2f qa = *(const v2f*)(tab + r);
    const v2f qb = *(const v2f*)(tab + r + 2);
    float s = qa.x * b1[0];
    s = fmaf(qa.y, b1[1], s);
    s = fmaf(qb.x, b1[2], s);
    s = fmaf(qb.y, b1[3], s);
    s = fmaf(tab[r + 4], b1[4], s);
    acc2 = b0[2] * s;
  }
  {  // row 3 (==3 mod 4): b32 + b64 + b64
    const int r = base + 3 * NGRID;
    const float f0 = tab[r];
    const v2f qa = *(const v2f*)(tab + r + 1);
    const v2f qb = *(const v2f*)(tab + r + 3);
    float s = f0 * b1[0];
    s = fmaf(qa.x, b1[1], s);
    s = fmaf(qa.y, b1[2], s);
    s = fmaf(qb.x, b1[3], s);
    s = fmaf(qb.y, b1[4], s);
    acc3 = b0[3] * s;
  }
  {  // row 4 (==0 mod 4): b128 + b32
    const int r = base + 4 * NGRID;
    const v4f q = *(const v4f*)(tab + r);
    float s = q.x * b1[0];
    s = fmaf(q.y, b1[1], s);
    s = fmaf(q.z, b1[2], s);
    s = fmaf(q.w, b1[3], s);
    s = fmaf(tab[r + 4], b1[4], s);
    acc4 = b0[4] * s;
  }
  return ((acc0 + acc1) + (acc2 + acc3)) + acc4;
}

__global__ __launch_bounds__(1024, 1)
void lagrange_interp_kernel(const float* __restrict__ inp,
                            const float* __restrict__ coe,
                            float* __restrict__ out,
                            int npts) {
  extern __shared__ float tab[];  // 258 KB coefficient table, whole grid
  const int tid = (int)threadIdx.x;
  const int nth = (int)blockDim.x;

  // ---- Stage the whole 257x257 table into LDS via the async copy engine ----
#if HAVE_ASYNC_B128
  for (int i = tid; i < NCHUNK; i += nth) {
    __builtin_amdgcn_global_load_async_to_lds_b128(
        (gptr_v4i)(coe + 4 * i),
        (lptr_v4i)(tab + 4 * i),
        /*offset=*/0, /*cpol=*/0);
  }
  if (tid == 0) tab[NTAB - 1] = coe[NTAB - 1];  // tail float
  #if HAVE_WAIT_ASYNC
  __builtin_amdgcn_s_wait_asynccnt(0);
  #else
  asm volatile("s_wait_asynccnt 0x0" ::: "memory");
  #endif
#else
  for (int i = tid; i < NTAB; i += nth) tab[i] = coe[i];
#endif
  __syncthreads();

  // ---- Two points per iteration: b128 in, 2x13 ds gathers, b64 out ----
  const v4f* __restrict__ in4 = (const v4f*)inp;   // (x0,y0,x1,y1)
  v2f* __restrict__ out2 = (v2f*)out;
  const int npairs = npts >> 1;
  const int gtid = (int)(blockIdx.x * blockDim.x) + tid;
  const int stride = (int)(gridDim.x * blockDim.x);

  for (int q = gtid; q < npairs; q += stride) {
    const v4f xy = __builtin_nontemporal_load(in4 + q);  // streaming, keep L2 for table
    __builtin_prefetch((const void*)(in4 + q + stride), 0, 0);

    float bA0[5], bA1[5], bB0[5], bB1[5];
    const int baseA = prep_point(xy.x, xy.y, bA0, bA1);
    const int baseB = prep_point(xy.z, xy.w, bB0, bB1);

    v2f r;
    r.x = gather_dot(tab, baseA, bA0, bA1);
    r.y = gather_dot(tab, baseB, bB0, bB1);
    __builtin_nontemporal_store(r, out2 + q);
  }

  // Odd tail (not hit for 2048x2048, kept for generality).
  for (int p = (npairs << 1) + gtid; p < npts; p += stride) {
    const float2 xy = ((const float2*)inp)[p];
    float b0[5], b1[5];
    const int base = prep_point(xy.x, xy.y, b0, b1);
    out[p] = gather_dot(tab, base, b0, b1);
  }
}

extern "C" void kernel_launch(void* const* d_in, const int* in_sizes, int n_in,
                              void* d_out, int out_size, void* d_ws, size_t ws_size,
                              hipStream_t stream) {
  (void)n_in; (void)d_ws; (void)ws_size; (void)out_size;
  const float* inp = (const float*)d_in[0];   // (2048,2048,2) f32
  const float* coe = (const float*)d_in[1];   // (257,257) f32
  float* out = (float*)d_out;                 // (2048,2048) f32
  const int npts = in_sizes[0] / 2;

  const int threads = 1024;                   // 32 wave32 waves -> fills a WGP
  const int blocks = 512;                     // ~4 pairs/thread grid-stride

  // 258 KB dynamic LDS > default cap: raise the attribute (host-side config,
  // not a stream op; safe under graph capture and deterministic).
  (void)hipFuncSetAttribute((const void*)lagrange_interp_kernel,
                            hipFuncAttributeMaxDynamicSharedMemorySize,
                            (int)TAB_BYTES);

  lagrange_interp_kernel<<<blocks, threads, TAB_BYTES, stream>>>(inp, coe, out, npts);
}